// MambaDecoder_2499670966925
// MI455X (gfx1250) — compile-verified
//
#include <hip/hip_runtime.h>
#include <hip/hip_bf16.h>

typedef float v2f __attribute__((ext_vector_type(2)));
typedef float v8f __attribute__((ext_vector_type(8)));

#define VOCAB 32000
#define DIM   512
#define NL    6
#define BB    2
#define LL    1024
#define DI    1024      // d_inner
#define DS    16        // d_state
#define DC    4         // d_conv
#define DR    32        // dt_rank
#define MROWS (BB*LL)   // 2048 token rows

#define WMMA4(A_, B_, C_) \
    __builtin_amdgcn_wmma_f32_16x16x4_f32(false, (A_), false, (B_), (short)0, (C_), false, false)

// ---------------------------------------------------------------------------
// x[b,l,:] = embed[tok[b,l],:] + pos_emb[0,l,:]
__global__ void embed_kernel(const int* __restrict__ tok,
                             const float* __restrict__ emb,
                             const float* __restrict__ pos,
                             float* __restrict__ x) {
    int idx = blockIdx.x * 256 + threadIdx.x;        // over MROWS*DIM
    int c = idx % DIM;
    int m = idx / DIM;
    int l = m % LL;
    x[idx] = emb[(size_t)tok[m] * DIM + c] + pos[l * DIM + c];
}

// ---------------------------------------------------------------------------
// fp32 WMMA GEMM: C[M,N] = A[M,K(lda)] * W[N,K]^T (+ bias[N])
// One wave -> 32x64 tile: 2 A-fragments x 4 B-fragments = 8 WMMAs / K-step,
// with explicit register double-buffering so the 6 loads of step k+4 are in
// flight while the 8 V_WMMA_F32_16X16X4_F32 of step k execute.
// f32 16x4 A fragment: VGPR0 = {K=k0 | K=k0+2}, VGPR1 = {k0+1 | k0+3}
// (lanes 0-15 | lanes 16-31); B fragment mirrors it with N across lanes.
__global__ void wmma_gemm_f32(const float* __restrict__ A,
                              const float* __restrict__ W,
                              float* __restrict__ C,
                              const float* __restrict__ bias,
                              int M, int N, int K, int lda) {
    int wave = (blockIdx.x * blockDim.x + threadIdx.x) >> 5;
    int lane = threadIdx.x & 31;
    int tilesM = M >> 5;                 // 32-row tiles
    int tilesN = N >> 6;                 // 64-col tiles
    if (wave >= tilesM * tilesN) return; // wave-uniform exit (EXEC stays full)
    int tm = wave % tilesM;
    int tn = wave / tilesM;
    int m0 = tm << 5;
    int n0 = tn << 6;
    int half = lane >> 4;                // 0: K=k0..k0+1, 1: K=k0+2..k0+3
    int lm   = lane & 15;
    int koff = half * 2;

    const float* Ap0 = A + (size_t)(m0 +  0 + lm) * lda + koff;
    const float* Ap1 = A + (size_t)(m0 + 16 + lm) * lda + koff;
    const float* W0  = W + (size_t)(n0 +  0 + lm) * K + koff;
    const float* W1  = W + (size_t)(n0 + 16 + lm) * K + koff;
    const float* W2  = W + (size_t)(n0 + 32 + lm) * K + koff;
    const float* W3  = W + (size_t)(n0 + 48 + lm) * K + koff;

    v8f acc0 = {}; v8f acc1 = {}; v8f acc2 = {}; v8f acc3 = {};
    v8f acc4 = {}; v8f acc5 = {}; v8f acc6 = {}; v8f acc7 = {};

    // prologue: load K-step 0
    v2f a0 = *(const v2f*)(Ap0);
    v2f a1 = *(const v2f*)(Ap1);
    v2f b0 = *(const v2f*)(W0);
    v2f b1 = *(const v2f*)(W1);
    v2f b2 = *(const v2f*)(W2);
    v2f b3 = *(const v2f*)(W3);

#pragma unroll 2
    for (int k0 = 0; k0 < K - 4; k0 += 4) {
        // issue next K-step's loads before consuming the current fragments
        v2f a0n = *(const v2f*)(Ap0 + k0 + 4);
        v2f a1n = *(const v2f*)(Ap1 + k0 + 4);
        v2f b0n = *(const v2f*)(W0 + k0 + 4);
        v2f b1n = *(const v2f*)(W1 + k0 + 4);
        v2f b2n = *(const v2f*)(W2 + k0 + 4);
        v2f b3n = *(const v2f*)(W3 + k0 + 4);

        acc0 = WMMA4(a0, b0, acc0);
        acc1 = WMMA4(a0, b1, acc1);
        acc2 = WMMA4(a0, b2, acc2);
        acc3 = WMMA4(a0, b3, acc3);
        acc4 = WMMA4(a1, b0, acc4);
        acc5 = WMMA4(a1, b1, acc5);
        acc6 = WMMA4(a1, b2, acc6);
        acc7 = WMMA4(a1, b3, acc7);

        a0 = a0n; a1 = a1n;
        b0 = b0n; b1 = b1n; b2 = b2n; b3 = b3n;
    }
    // epilogue K-step
    acc0 = WMMA4(a0, b0, acc0);
    acc1 = WMMA4(a0, b1, acc1);
    acc2 = WMMA4(a0, b2, acc2);
    acc3 = WMMA4(a0, b3, acc3);
    acc4 = WMMA4(a1, b0, acc4);
    acc5 = WMMA4(a1, b1, acc5);
    acc6 = WMMA4(a1, b2, acc6);
    acc7 = WMMA4(a1, b3, acc7);

    float bv0 = bias ? bias[n0 +  0 + lm] : 0.0f;
    float bv1 = bias ? bias[n0 + 16 + lm] : 0.0f;
    float bv2 = bias ? bias[n0 + 32 + lm] : 0.0f;
    float bv3 = bias ? bias[n0 + 48 + lm] : 0.0f;

    // C/D layout: VGPR v -> row base+v (lanes 0-15) / base+8+v (lanes 16-31)
    int crow0 = m0 + half * 8;
    int crow1 = m0 + 16 + half * 8;
#pragma unroll
    for (int v = 0; v < 8; ++v) {
        float* cp0 = C + (size_t)(crow0 + v) * N + n0 + lm;
        cp0[ 0] = acc0[v] + bv0;
        cp0[16] = acc1[v] + bv1;
        cp0[32] = acc2[v] + bv2;
        cp0[48] = acc3[v] + bv3;
        float* cp1 = C + (size_t)(crow1 + v) * N + n0 + lm;
        cp1[ 0] = acc4[v] + bv0;
        cp1[16] = acc5[v] + bv1;
        cp1[32] = acc6[v] + bv2;
        cp1[48] = acc7[v] + bv3;
    }
}

// ---------------------------------------------------------------------------
__device__ __forceinline__ float silu_f(float v) {
    return v / (1.0f + __expf(-v));
}

// xc = silu(causal depthwise conv(xz[:, :DI])) ; xz row stride = 2*DI
__global__ void conv_silu_kernel(const float* __restrict__ xz,
                                 const float* __restrict__ cw,
                                 const float* __restrict__ cb,
                                 float* __restrict__ xc) {
    int idx = blockIdx.x * 256 + threadIdx.x;        // over MROWS*DI
    int d = idx % DI;
    int m = idx / DI;
    int l = m % LL;
    const float* base = xz + (size_t)m * (2 * DI) + d;
    float y = cb[d];
#pragma unroll
    for (int k = 0; k < DC; ++k) {
        int j = l - (DC - 1) + k;                    // causal left pad
        if (j >= 0) y += cw[d * DC + k] * base[(long)(k - (DC - 1)) * (2 * DI)];
    }
    xc[idx] = silu_f(y);
}

// in-place: buf = softplus(buf + dtb[d])
__global__ void softplus_bias_kernel(float* __restrict__ buf,
                                     const float* __restrict__ dtb) {
    int idx = blockIdx.x * 256 + threadIdx.x;        // over MROWS*DI
    int d = idx % DI;
    float v = buf[idx] + dtb[d];
    buf[idx] = (v > 20.0f) ? v : log1pf(__expf(v));
}

// selective scan: one lane per (b, d, n); 16-lane shfl_xor reduction over n
__global__ void scan_kernel(const float* __restrict__ xc,
                            const float* __restrict__ dt,
                            const float* __restrict__ xdbl,
                            const float* __restrict__ A_log,
                            const float* __restrict__ Dp,
                            float* __restrict__ y) {
    int g = blockIdx.x * 256 + threadIdx.x;          // over BB*DI*DS = 32768
    int n = g & (DS - 1);
    int d = (g >> 4) % DI;
    int b = g / (DI * DS);

    float a  = -__expf(A_log[d * DS + n]);           // A = -exp(A_log)
    float Dv = Dp[d];
    float h  = 0.0f;

    const float* dtp = dt   + (size_t)b * LL * DI + d;
    const float* up  = xc   + (size_t)b * LL * DI + d;
    const float* xdp = xdbl + (size_t)b * LL * 64;
    float*       yp  = y    + (size_t)b * LL * DI + d;

    for (int l = 0; l < LL; ++l) {
        float dtv = dtp[(size_t)l * DI];
        float u   = up[(size_t)l * DI];
        float bn  = xdp[l * 64 + DR + n];            // B block at cols [32,48)
        float cn  = xdp[l * 64 + DR + DS + n];       // C block at cols [48,64)
        h = __expf(dtv * a) * h + dtv * bn * u;
        float part = h * cn;
        part += __shfl_xor(part, 1);
        part += __shfl_xor(part, 2);
        part += __shfl_xor(part, 4);
        part += __shfl_xor(part, 8);
        if (n == 0) yp[(size_t)l * DI] = part + u * Dv;
    }
}

// y *= silu(z), z = xz[:, DI:2*DI]
__global__ void gate_kernel(const float* __restrict__ xz,
                            float* __restrict__ y) {
    int idx = blockIdx.x * 256 + threadIdx.x;        // over MROWS*DI
    int d = idx % DI;
    int m = idx / DI;
    float z = xz[(size_t)m * (2 * DI) + DI + d];
    y[idx] *= silu_f(z);
}

// x = LayerNorm(x + blk) * g + b   (one block per token row, biased variance)
__global__ void add_ln_kernel(const float* __restrict__ blk,
                              float* __restrict__ x,
                              const float* __restrict__ g,
                              const float* __restrict__ b) {
    __shared__ float ssum[256];
    __shared__ float ssq[256];
    int m = blockIdx.x;
    int t = threadIdx.x;
    float v0 = x[(size_t)m * DIM + t]       + blk[(size_t)m * DIM + t];
    float v1 = x[(size_t)m * DIM + t + 256] + blk[(size_t)m * DIM + t + 256];
    ssum[t] = v0 + v1;
    ssq[t]  = v0 * v0 + v1 * v1;
    __syncthreads();
    for (int off = 128; off > 0; off >>= 1) {
        if (t < off) { ssum[t] += ssum[t + off]; ssq[t] += ssq[t + off]; }
        __syncthreads();
    }
    float mu  = ssum[0] * (1.0f / DIM);
    float var = ssq[0] * (1.0f / DIM) - mu * mu;
    float rs  = rsqrtf(var + 1e-5f);
    x[(size_t)m * DIM + t]       = (v0 - mu) * rs * g[t]       + b[t];
    x[(size_t)m * DIM + t + 256] = (v1 - mu) * rs * g[t + 256] + b[t + 256];
}

// ---------------------------------------------------------------------------
extern "C" void kernel_launch(void* const* d_in, const int* in_sizes, int n_in,
                              void* d_out, int out_size, void* d_ws, size_t ws_size,
                              hipStream_t stream) {
    (void)in_sizes; (void)n_in; (void)out_size; (void)ws_size;
    const int*   tokens = (const int*)  d_in[0];
    const float* pos    = (const float*)d_in[1];
    const float* emb    = (const float*)d_in[2];
    const float* lm_w   = (const float*)d_in[3];
    const float* lm_b   = (const float*)d_in[4];
    const float* in_w   = (const float*)d_in[5];
    const float* cw     = (const float*)d_in[6];
    const float* cb     = (const float*)d_in[7];
    const float* xp_w   = (const float*)d_in[8];
    const float* dtw    = (const float*)d_in[9];
    const float* dtb    = (const float*)d_in[10];
    const float* A_log  = (const float*)d_in[11];
    const float* Dp     = (const float*)d_in[12];
    const float* ow     = (const float*)d_in[13];
    const float* lng    = (const float*)d_in[14];
    const float* lnb    = (const float*)d_in[15];
    float* out = (float*)d_out;

    // workspace layout (floats): ~48.5 MB total
    float* ws    = (float*)d_ws;
    float* x     = ws;                                // MROWS*DIM
    float* xz    = x     + (size_t)MROWS * DIM;       // MROWS*2*DI
    float* xc    = xz    + (size_t)MROWS * 2 * DI;    // MROWS*DI
    float* xdbl  = xc    + (size_t)MROWS * DI;        // MROWS*64
    float* dtbuf = xdbl  + (size_t)MROWS * 64;        // MROWS*DI
    float* ybuf  = dtbuf + (size_t)MROWS * DI;        // MROWS*DI
    float* blk   = ybuf  + (size_t)MROWS * DI;        // MROWS*DIM

    embed_kernel<<<(MROWS * DIM) / 256, 256, 0, stream>>>(tokens, emb, pos, x);

    for (int i = 0; i < NL; ++i) {
        const float* in_wi = in_w + (size_t)i * 2 * DI * DIM;
        const float* cwi   = cw   + (size_t)i * DI * DC;
        const float* cbi   = cb   + (size_t)i * DI;
        const float* xpwi  = xp_w + (size_t)i * (DR + 2 * DS) * DI;
        const float* dtwi  = dtw  + (size_t)i * DI * DR;
        const float* dtbi  = dtb  + (size_t)i * DI;
        const float* Ali   = A_log+ (size_t)i * DI * DS;
        const float* Dpi   = Dp   + (size_t)i * DI;
        const float* owi   = ow   + (size_t)i * DIM * DI;
        const float* lngi  = lng  + (size_t)i * DIM;
        const float* lnbi  = lnb  + (size_t)i * DIM;

        // in_proj: (2048x512) x (2048x512)^T -> (2048x2048)
        wmma_gemm_f32<<<((MROWS / 32) * (2 * DI / 64)) / 8, 256, 0, stream>>>(
            x, in_wi, xz, nullptr, MROWS, 2 * DI, DIM, DIM);

        conv_silu_kernel<<<(MROWS * DI) / 256, 256, 0, stream>>>(xz, cwi, cbi, xc);

        // x_proj: (2048x1024) x (64x1024)^T -> (2048x64)
        wmma_gemm_f32<<<((MROWS / 32) * (64 / 64)) / 8, 256, 0, stream>>>(
            xc, xpwi, xdbl, nullptr, MROWS, 64, DI, DI);

        // dt_proj: A = xdbl[:, 0:32] (lda=64), W = (1024x32) -> (2048x1024)
        wmma_gemm_f32<<<((MROWS / 32) * (DI / 64)) / 8, 256, 0, stream>>>(
            xdbl, dtwi, dtbuf, nullptr, MROWS, DI, DR, 64);

        softplus_bias_kernel<<<(MROWS * DI) / 256, 256, 0, stream>>>(dtbuf, dtbi);

        scan_kernel<<<(BB * DI * DS) / 256, 256, 0, stream>>>(
            xc, dtbuf, xdbl, Ali, Dpi, ybuf);

        gate_kernel<<<(MROWS * DI) / 256, 256, 0, stream>>>(xz, ybuf);

        // out_proj: (2048x1024) x (512x1024)^T -> (2048x512)
        wmma_gemm_f32<<<((MROWS / 32) * (DIM / 64)) / 8, 256, 0, stream>>>(
            ybuf, owi, blk, nullptr, MROWS, DIM, DI, DI);

        add_ln_kernel<<<MROWS, 256, 0, stream>>>(blk, x, lngi, lnbi);
    }

    // lm_head: (2048x512) x (32000x512)^T + lm_b -> (2048x32000)
    wmma_gemm_f32<<<((MROWS / 32) * (VOCAB / 64)) / 8, 256, 0, stream>>>(
        x, lm_w, out, lm_b, MROWS, VOCAB, DIM, DIM);
}